// DA_HGNN_78417512891092
// MI455X (gfx1250) — compile-verified
//
#include <hip/hip_runtime.h>
#include <math.h>

typedef __attribute__((ext_vector_type(2))) float v2f;
typedef __attribute__((ext_vector_type(8))) float v8f;

constexpr int   N_    = 4096;
constexpr int   FIN   = 784;
constexpr int   D_    = 256;
constexpr int   KNN   = 11;          // top_k + 1
constexpr float DV    = 0.30151134457776363f;  // 11^-0.5 (row sums of H are always 11)
constexpr float SIGMA = 0.3f;
constexpr float SLOPE = 0.2f;
constexpr float EPSV  = 1e-6f;

__device__ __forceinline__ float lrelu(float x) { return x > 0.f ? x : SLOPE * x; }

// ---------------------------------------------------------------------------
// WMMA f32 GEMM:  C[M,N] = A[M,K] @ B  (B row-major [K,N] or, if transB, [N,K])
// block = 256 threads (8 waves), block tile 64x128, wave tile 32x32 (2x2 WMMA)
// Requires M%64==0, N%128==0, K%16==0 (true for all shapes used here).
// ---------------------------------------------------------------------------
__global__ __launch_bounds__(256)
void gemm_wmma_f32(const float* __restrict__ A, const float* __restrict__ B,
                   float* __restrict__ C, int M, int N, int Kd,
                   int lda, int ldb, int ldc, int transB, int applyElu)
{
    __shared__ float As[16][72];   // transposed [k][m], padded
    __shared__ float Bs[16][128];  // [k][n]

    const int tid  = threadIdx.x;
    const int bm   = blockIdx.y * 64;
    const int bn   = blockIdx.x * 128;
    const int wave = tid >> 5;
    const int lane = tid & 31;
    const int wm   = (wave & 1) * 32;
    const int wn   = (wave >> 1) * 32;
    const int half = lane >> 4;
    const int l16  = lane & 15;

    v8f acc[2][2];
    acc[0][0] = {}; acc[0][1] = {}; acc[1][0] = {}; acc[1][1] = {};

    for (int k0 = 0; k0 < Kd; k0 += 16) {
        // stage A tile 64x16 -> As[k][m]   (coalesced global reads)
        for (int i = tid; i < 1024; i += 256) {
            int m = i >> 4, kk = i & 15;
            As[kk][m] = A[(bm + m) * lda + k0 + kk];
        }
        // stage B tile 16x128 -> Bs[k][n]
        for (int i = tid; i < 2048; i += 256) {
            int kk = i >> 7, n = i & 127;
            Bs[kk][n] = transB ? B[(bn + n) * ldb + k0 + kk]
                               : B[(k0 + kk) * ldb + bn + n];
        }
        // prefetch next K-tile into cache while this tile computes
        if (k0 + 16 < Kd) {
            if (tid < 64)
                __builtin_prefetch(&A[(bm + tid) * lda + k0 + 16], 0, 1);
            if (transB) {
                if (tid >= 64 && tid < 192)
                    __builtin_prefetch(&B[(bn + tid - 64) * ldb + k0 + 16], 0, 1);
            } else {
                if (tid >= 64 && tid < 80)
                    __builtin_prefetch(&B[(k0 + 16 + tid - 64) * ldb + bn], 0, 1);
            }
        }
        __syncthreads();

        #pragma unroll
        for (int ks = 0; ks < 16; ks += 4) {
            v2f a[2], b[2];
            #pragma unroll
            for (int t = 0; t < 2; ++t) {
                // A frag (16x4): lanes0-15 hold K=0,1; lanes16-31 hold K=2,3
                int mr = wm + t * 16 + l16;
                a[t].x = As[ks + half * 2 + 0][mr];
                a[t].y = As[ks + half * 2 + 1][mr];
                // B frag (4x16): same K split across lane halves
                int nc = wn + t * 16 + l16;
                b[t].x = Bs[ks + half * 2 + 0][nc];
                b[t].y = Bs[ks + half * 2 + 1][nc];
            }
            #pragma unroll
            for (int tm = 0; tm < 2; ++tm)
                #pragma unroll
                for (int tn = 0; tn < 2; ++tn)
                    acc[tm][tn] = __builtin_amdgcn_wmma_f32_16x16x4_f32(
                        false, a[tm], false, b[tn], (short)0, acc[tm][tn], false, false);
        }
        __syncthreads();
    }

    // C layout: VGPR r -> rows r (lanes0-15) / r+8 (lanes16-31), N = l16
    #pragma unroll
    for (int tm = 0; tm < 2; ++tm)
        #pragma unroll
        for (int tn = 0; tn < 2; ++tn)
            #pragma unroll
            for (int r = 0; r < 8; ++r) {
                int row = bm + wm + tm * 16 + half * 8 + r;
                int col = bn + wn + tn * 16 + l16;
                float v = acc[tm][tn][r];
                if (applyElu) v = v > 0.f ? v : expm1f(v);
                C[row * ldc + col] = v;
            }
}

// ---------------------------------------------------------------------------
// Fused attention GEMM:  C = elu( COE.T @ B ),  COE.T[j,i] generated on the
// fly from the rank-1 attention structure:
//     A[j,i] = exp(leaky(u[i] + v[j]) + rt[i]) / den[j]
// This removes the 64MB P materialization (256MB of HBM traffic per GEMM).
// Same tiling as gemm_wmma_f32; B is row-major [K,N] (NN only).
// ---------------------------------------------------------------------------
__global__ __launch_bounds__(256)
void gemm_attn_wmma(const float* __restrict__ u, const float* __restrict__ v,
                    const float* __restrict__ rt, const float* __restrict__ den,
                    const float* __restrict__ B, float* __restrict__ C,
                    int M, int N, int Kd, int ldb, int ldc)
{
    __shared__ float As[16][72];
    __shared__ float Bs[16][128];
    __shared__ float vs[64];    // v[bm+m]
    __shared__ float dens[64];  // den[bm+m]

    const int tid  = threadIdx.x;
    const int bm   = blockIdx.y * 64;
    const int bn   = blockIdx.x * 128;
    const int wave = tid >> 5;
    const int lane = tid & 31;
    const int wm   = (wave & 1) * 32;
    const int wn   = (wave >> 1) * 32;
    const int half = lane >> 4;
    const int l16  = lane & 15;

    if (tid < 64) { vs[tid] = v[bm + tid]; dens[tid] = den[bm + tid]; }
    __syncthreads();

    v8f acc[2][2];
    acc[0][0] = {}; acc[0][1] = {}; acc[1][0] = {}; acc[1][1] = {};

    for (int k0 = 0; k0 < Kd; k0 += 16) {
        // generate A tile procedurally (u/rt are L1-resident 16KB vectors)
        for (int i = tid; i < 1024; i += 256) {
            int m = i >> 4, kk = i & 15;
            float ui = u[k0 + kk], ri = rt[k0 + kk];
            As[kk][m] = expf(lrelu(ui + vs[m]) + ri) / dens[m];
        }
        for (int i = tid; i < 2048; i += 256) {
            int kk = i >> 7, n = i & 127;
            Bs[kk][n] = B[(k0 + kk) * ldb + bn + n];
        }
        if (k0 + 16 < Kd && tid >= 64 && tid < 80)
            __builtin_prefetch(&B[(k0 + 16 + tid - 64) * ldb + bn], 0, 1);
        __syncthreads();

        #pragma unroll
        for (int ks = 0; ks < 16; ks += 4) {
            v2f a[2], b[2];
            #pragma unroll
            for (int t = 0; t < 2; ++t) {
                int mr = wm + t * 16 + l16;
                a[t].x = As[ks + half * 2 + 0][mr];
                a[t].y = As[ks + half * 2 + 1][mr];
                int nc = wn + t * 16 + l16;
                b[t].x = Bs[ks + half * 2 + 0][nc];
                b[t].y = Bs[ks + half * 2 + 1][nc];
            }
            #pragma unroll
            for (int tm = 0; tm < 2; ++tm)
                #pragma unroll
                for (int tn = 0; tn < 2; ++tn)
                    acc[tm][tn] = __builtin_amdgcn_wmma_f32_16x16x4_f32(
                        false, a[tm], false, b[tn], (short)0, acc[tm][tn], false, false);
        }
        __syncthreads();
    }

    #pragma unroll
    for (int tm = 0; tm < 2; ++tm)
        #pragma unroll
        for (int tn = 0; tn < 2; ++tn)
            #pragma unroll
            for (int r = 0; r < 8; ++r) {
                int row = bm + wm + tm * 16 + half * 8 + r;
                int col = bn + wn + tn * 16 + l16;
                float x = acc[tm][tn][r];
                C[row * ldc + col] = x > 0.f ? x : expm1f(x);   // ELU
            }
}

// ---------------------------------------------------------------------------
// sqv[i] = G[i,i]   (squared norms of X rows, read off the Gram diagonal)
// ---------------------------------------------------------------------------
__global__ void diag_kernel(const float* __restrict__ G, float* __restrict__ sqv)
{
    int i = blockIdx.x * 256 + threadIdx.x;
    if (i < N_) sqv[i] = G[(size_t)i * N_ + i];
}

// ---------------------------------------------------------------------------
// Per-row top-KNN smallest of dist[j] = |sq[j] + sq[i] - 2 G[i,j]|
// Lexicographic (value, index) ordering matches lax.top_k tie-breaking.
// ---------------------------------------------------------------------------
__device__ __forceinline__ void tk_ins(float d, int j, float (&bv)[KNN], int (&bi)[KNN])
{
    if (d < bv[KNN - 1] || (d == bv[KNN - 1] && j < bi[KNN - 1])) {
        bv[KNN - 1] = d; bi[KNN - 1] = j;
        #pragma unroll
        for (int t = KNN - 1; t > 0; --t) {
            bool sw = (bv[t] < bv[t - 1]) || (bv[t] == bv[t - 1] && bi[t] < bi[t - 1]);
            if (sw) {
                float tv = bv[t]; bv[t] = bv[t - 1]; bv[t - 1] = tv;
                int   ti = bi[t]; bi[t] = bi[t - 1]; bi[t - 1] = ti;
            }
        }
    }
}

__global__ __launch_bounds__(256)
void topk_kernel(const float* __restrict__ G, const float* __restrict__ sqv,
                 int* __restrict__ idx)
{
    __shared__ float sv[256 * KNN];
    __shared__ int   si[256 * KNN];
    const int row = blockIdx.x;
    const int tid = threadIdx.x;
    const float sqi = sqv[row];

    float bv[KNN]; int bi[KNN];
    #pragma unroll
    for (int t = 0; t < KNN; ++t) { bv[t] = 3.4e38f; bi[t] = 0x7fffffff; }

    for (int j = tid; j < N_; j += 256) {
        float d = fabsf(sqv[j] + sqi - 2.f * G[(size_t)row * N_ + j]);
        tk_ins(d, j, bv, bi);
    }
    #pragma unroll
    for (int t = 0; t < KNN; ++t) { sv[tid * KNN + t] = bv[t]; si[tid * KNN + t] = bi[t]; }
    __syncthreads();

    if (tid < 32) {  // stage 1: each of 32 threads merges 8 lists
        float mv[KNN]; int mi[KNN];
        #pragma unroll
        for (int t = 0; t < KNN; ++t) { mv[t] = 3.4e38f; mi[t] = 0x7fffffff; }
        for (int s = 0; s < 8; ++s) {
            int base = (tid * 8 + s) * KNN;
            #pragma unroll
            for (int t = 0; t < KNN; ++t) tk_ins(sv[base + t], si[base + t], mv, mi);
        }
        int ob = (tid * 8) * KNN;
        #pragma unroll
        for (int t = 0; t < KNN; ++t) { sv[ob + t] = mv[t]; si[ob + t] = mi[t]; }
    }
    __syncthreads();

    if (tid == 0) {  // stage 2: merge the 32 survivors
        float mv[KNN]; int mi[KNN];
        #pragma unroll
        for (int t = 0; t < KNN; ++t) { mv[t] = 3.4e38f; mi[t] = 0x7fffffff; }
        for (int g = 0; g < 32; ++g) {
            int base = (g * 8) * KNN;
            #pragma unroll
            for (int t = 0; t < KNN; ++t) tk_ins(sv[base + t], si[base + t], mv, mi);
        }
        #pragma unroll
        for (int t = 0; t < KNN; ++t) idx[row * KNN + t] = mi[t];
    }
}

// ---------------------------------------------------------------------------
// CSC build for H (columns = hyperedges)
// ---------------------------------------------------------------------------
__global__ void colcount_kernel(const int* __restrict__ idx, int* __restrict__ colcnt)
{
    int e = blockIdx.x * 256 + threadIdx.x;
    if (e < N_ * KNN) atomicAdd(&colcnt[idx[e]], 1);
}

__global__ void scan_kernel(const int* __restrict__ colcnt, int* __restrict__ colptr,
                            float* __restrict__ de)
{
    int acc = 0;
    for (int k = 0; k < N_; ++k) {
        colptr[k] = acc;
        de[k] = rsqrtf((float)colcnt[k]);   // De^{-1/2}
        acc += colcnt[k];
    }
    colptr[N_] = acc;
}

__global__ void colfill_kernel(const int* __restrict__ idx, const int* __restrict__ colptr,
                               int* __restrict__ colfill, int* __restrict__ colidx)
{
    int e = blockIdx.x * 256 + threadIdx.x;
    if (e < N_ * KNN) {
        int k = idx[e];
        int p = atomicAdd(&colfill[k], 1);
        colidx[colptr[k] + p] = e / KNN;
    }
}

// tmp[k,:] = de[k] * dv * sum_{i in col(k)} X[i,:]      (H.T @ (dv*X), scaled)
__global__ __launch_bounds__(256)
void gather_cols_kernel(const float* __restrict__ X, const int* __restrict__ colptr,
                        const int* __restrict__ colidx, const float* __restrict__ de,
                        float* __restrict__ tmp)
{
    int k = blockIdx.x;
    int beg = colptr[k], end = colptr[k + 1];
    float s = de[k] * DV;
    for (int f = threadIdx.x; f < FIN; f += 256) {
        float a = 0.f;
        for (int m = beg; m < end; ++m) a += X[colidx[m] * FIN + f];
        tmp[k * FIN + f] = s * a;
    }
}

// E[i,:] = dv * sum_{k in idx[i]} de[k] * Xl[k,:]
__global__ __launch_bounds__(256)
void row_gather_kernel(const float* __restrict__ Xl, const int* __restrict__ idx,
                       const float* __restrict__ de, float* __restrict__ E)
{
    int i = blockIdx.x;
    int f = threadIdx.x;
    float a = 0.f;
    #pragma unroll
    for (int t = 0; t < KNN; ++t) {
        int k = idx[i * KNN + t];
        a += de[k] * Xl[k * D_ + f];
    }
    E[i * D_ + f] = DV * a;
}

// norms[i] = ||Xl[i,:]||
__global__ __launch_bounds__(256)
void norms_kernel(const float* __restrict__ Xl, float* __restrict__ norms)
{
    __shared__ float red[256];
    int i = blockIdx.x, t = threadIdx.x;
    float v = Xl[i * D_ + t];
    red[t] = v * v;
    __syncthreads();
    for (int s = 128; s > 0; s >>= 1) { if (t < s) red[t] += red[t + s]; __syncthreads(); }
    if (t == 0) norms[i] = sqrtf(red[0]);
}

// rho_x[i] = sum_j { j!=i, cos(Xl_i,Xl_j) > sigma, idx[i] ∩ idx[j] != ∅ } cos(...)
__global__ __launch_bounds__(256)
void rho_kernel(const float* __restrict__ Snum, const float* __restrict__ norms,
                const int* __restrict__ idx, float* __restrict__ rho_x)
{
    __shared__ int   myi[KNN];
    __shared__ float red[256];
    const int i = blockIdx.x, tid = threadIdx.x;
    if (tid < KNN) myi[tid] = idx[i * KNN + tid];
    __syncthreads();

    const float ni = norms[i];
    float accum = 0.f;
    for (int j = tid; j < N_; j += 256) {
        if (j == i) continue;
        float s = Snum[(size_t)i * N_ + j] / fmaxf(ni * norms[j], EPSV);
        if (s <= SIGMA) continue;
        int oj[KNN];
        #pragma unroll
        for (int b = 0; b < KNN; ++b) oj[b] = idx[j * KNN + b];
        bool adj = false;
        #pragma unroll
        for (int a = 0; a < KNN; ++a) {
            int v = myi[a];
            #pragma unroll
            for (int b = 0; b < KNN; ++b) adj |= (v == oj[b]);
        }
        if (adj) accum += s;
    }
    red[tid] = accum;
    __syncthreads();
    for (int s = 128; s > 0; s >>= 1) { if (tid < s) red[tid] += red[tid + s]; __syncthreads(); }
    if (tid == 0) rho_x[i] = red[0];
}

// rho_e[k] += rho_x[i] for each k in idx[i]
__global__ void rho_e_kernel(const int* __restrict__ idx, const float* __restrict__ rho_x,
                             float* __restrict__ rho_e)
{
    int e = blockIdx.x * 256 + threadIdx.x;
    if (e < N_ * KNN) atomicAdd(&rho_e[idx[e]], rho_x[e / KNN]);
}

// fused rank-1 attention vectors:
//   u_x = (Xl@W)·ax_lo, v_x = (E@W)·ax_hi, u_e = (E@W)·ae_lo, v_e = (Xl@W)·ae_hi
__global__ __launch_bounds__(256)
void uv_kernel(const float* __restrict__ WXl, const float* __restrict__ WE,
               const float* __restrict__ ax, const float* __restrict__ ae,
               float* __restrict__ u_x, float* __restrict__ v_x,
               float* __restrict__ u_e, float* __restrict__ v_e)
{
    __shared__ float red[256];
    const int i = blockIdx.x, t = threadIdx.x;
    const float wxl = WXl[i * D_ + t];
    const float we  = WE[i * D_ + t];

    float p[4] = { wxl * ax[t], we * ax[D_ + t], we * ae[t], wxl * ae[D_ + t] };
    float* outs[4] = { u_x, v_x, u_e, v_e };
    for (int q = 0; q < 4; ++q) {
        red[t] = p[q];
        __syncthreads();
        for (int s = 128; s > 0; s >>= 1) { if (t < s) red[t] += red[t + s]; __syncthreads(); }
        if (t == 0) outs[q][i] = red[0];
        __syncthreads();
    }
}

// 6 max reductions: u_x, v_x, u_e, v_e, rho_x, rho_e -> scal[0..5]
__global__ __launch_bounds__(256)
void max6_kernel(const float* __restrict__ u_x, const float* __restrict__ v_x,
                 const float* __restrict__ u_e, const float* __restrict__ v_e,
                 const float* __restrict__ rho_x, const float* __restrict__ rho_e,
                 float* __restrict__ scal)
{
    __shared__ float red[256];
    const float* src;
    switch (blockIdx.x) {
        case 0: src = u_x; break;  case 1: src = v_x; break;
        case 2: src = u_e; break;  case 3: src = v_e; break;
        case 4: src = rho_x; break; default: src = rho_e; break;
    }
    int t = threadIdx.x;
    float m = -3.4e38f;
    for (int i = t; i < N_; i += 256) m = fmaxf(m, src[i]);
    red[t] = m;
    __syncthreads();
    for (int s = 128; s > 0; s >>= 1) { if (t < s) red[t] = fmaxf(red[t], red[t + s]); __syncthreads(); }
    if (t == 0) scal[blockIdx.x] = red[0];
}

// rho_t = rho / max(rho) * max(a);  max(a) = leaky(max u + max v) (rank-1 + monotone)
__global__ void rho_t_kernel(const float* __restrict__ rho_x, const float* __restrict__ rho_e,
                             const float* __restrict__ scal,
                             float* __restrict__ rt_x, float* __restrict__ rt_e)
{
    int i = blockIdx.x * 256 + threadIdx.x;
    if (i < N_) {
        float max_ax = lrelu(scal[0] + scal[1]);
        float max_ae = lrelu(scal[2] + scal[3]);
        rt_x[i] = rho_x[i] / scal[4] * max_ax;
        rt_e[i] = rho_e[i] / scal[5] * max_ae;
    }
}

// den[j] = sum_{i in col(j)} exp(leaky(u_i+v_j) + rho_t_i)   (the _coef denominator)
__global__ void den_kernel(const int* __restrict__ colptr, const int* __restrict__ colidx,
                           const float* __restrict__ u_x, const float* __restrict__ v_x,
                           const float* __restrict__ rt_x,
                           const float* __restrict__ u_e, const float* __restrict__ v_e,
                           const float* __restrict__ rt_e,
                           float* __restrict__ den_x, float* __restrict__ den_e)
{
    int j = blockIdx.x * 256 + threadIdx.x;
    if (j >= N_) return;
    float vx = v_x[j], ve = v_e[j];
    float dx = 0.f, dn = 0.f;
    int beg = colptr[j], end = colptr[j + 1];
    for (int m = beg; m < end; ++m) {
        int i = colidx[m];
        dx += expf(lrelu(u_x[i] + vx) + rt_x[i]);
        dn += expf(lrelu(u_e[i] + ve) + rt_e[i]);
    }
    den_x[j] = dx;
    den_e[j] = dn;
}

// ---------------------------------------------------------------------------
extern "C" void kernel_launch(void* const* d_in, const int* in_sizes, int n_in,
                              void* d_out, int out_size, void* d_ws, size_t ws_size,
                              hipStream_t stream)
{
    const float* X     = (const float*)d_in[0];  // [4096,784]
    const float* theta = (const float*)d_in[1];  // [784,256]
    const float* W     = (const float*)d_in[2];  // [256,256]
    const float* ax    = (const float*)d_in[3];  // [512]
    const float* ae    = (const float*)d_in[4];  // [512]
    float* out = (float*)d_out;                  // [4096,256]

    char* base = (char*)d_ws;
    size_t off = 0;
    auto alloc = [&](size_t bytes) -> void* {
        off = (off + 255) & ~(size_t)255;
        void* p = base + off;
        off += bytes;
        return p;
    };

    float* big     = (float*)alloc((size_t)N_ * N_ * 4);     // G / Snum
    float* tmp     = (float*)alloc((size_t)N_ * FIN * 4);
    float* Xl      = (float*)alloc((size_t)N_ * D_ * 4);
    float* E       = (float*)alloc((size_t)N_ * D_ * 4);
    float* WXl     = (float*)alloc((size_t)N_ * D_ * 4);
    float* WE      = (float*)alloc((size_t)N_ * D_ * 4);
    float* Etil    = (float*)alloc((size_t)N_ * D_ * 4);
    int*   idx     = (int*)alloc((size_t)N_ * KNN * 4);
    int*   colcnt  = (int*)alloc((size_t)N_ * 4);
    int*   colptr  = (int*)alloc((size_t)(N_ + 1) * 4);
    int*   colfil  = (int*)alloc((size_t)N_ * 4);
    int*   colidx  = (int*)alloc((size_t)N_ * KNN * 4);
    float* de      = (float*)alloc((size_t)N_ * 4);
    float* sqv     = (float*)alloc((size_t)N_ * 4);
    float* norms   = (float*)alloc((size_t)N_ * 4);
    float* u_x     = (float*)alloc((size_t)N_ * 4);
    float* v_x     = (float*)alloc((size_t)N_ * 4);
    float* u_e     = (float*)alloc((size_t)N_ * 4);
    float* v_e     = (float*)alloc((size_t)N_ * 4);
    float* rho_x   = (float*)alloc((size_t)N_ * 4);
    float* rho_e   = (float*)alloc((size_t)N_ * 4);
    float* rt_x    = (float*)alloc((size_t)N_ * 4);
    float* rt_e    = (float*)alloc((size_t)N_ * 4);
    float* den_x   = (float*)alloc((size_t)N_ * 4);
    float* den_e   = (float*)alloc((size_t)N_ * 4);
    float* scal    = (float*)alloc(64);

    const dim3 blk(256);
    const int NE = N_ * KNN;                  // 45056, multiple of 256
    const dim3 grid_big(N_ / 128, N_ / 64);   // (32,64)
    const dim3 grid_nar(D_ / 128, N_ / 64);   // (2,64)

    // 1. G = X @ X.T   (NT, K=784)
    gemm_wmma_f32<<<grid_big, blk, 0, stream>>>(X, X, big, N_, N_, FIN, FIN, FIN, N_, 1, 0);
    // 2. diag -> squared norms
    diag_kernel<<<N_ / 256, blk, 0, stream>>>(big, sqv);
    // 3. per-row top-11 smallest distances -> idx (the hyperedges of each node)
    topk_kernel<<<N_, blk, 0, stream>>>(big, sqv, idx);
    // 4. CSC build for H
    hipMemsetAsync(colcnt, 0, (size_t)N_ * 4, stream);
    hipMemsetAsync(colfil, 0, (size_t)N_ * 4, stream);
    hipMemsetAsync(rho_e, 0, (size_t)N_ * 4, stream);
    colcount_kernel<<<NE / 256, blk, 0, stream>>>(idx, colcnt);
    scan_kernel<<<1, 1, 0, stream>>>(colcnt, colptr, de);
    colfill_kernel<<<NE / 256, blk, 0, stream>>>(idx, colptr, colfil, colidx);
    // 5. tmp = De^{-1/2} (H.T @ (Dv^{-1/2} X))
    gather_cols_kernel<<<N_, blk, 0, stream>>>(X, colptr, colidx, de, tmp);
    // 6. Xl = tmp @ theta   (NN, K=784)
    gemm_wmma_f32<<<grid_nar, blk, 0, stream>>>(tmp, theta, Xl, N_, D_, FIN, FIN, D_, D_, 0, 0);
    // 7. E = Dv^{-1/2} (H @ (De^{-1/2} Xl))
    row_gather_kernel<<<N_, blk, 0, stream>>>(Xl, idx, de, E);
    // 8. Snum = Xl @ Xl.T (reuses big)
    gemm_wmma_f32<<<grid_big, blk, 0, stream>>>(Xl, Xl, big, N_, N_, D_, D_, D_, N_, 1, 0);
    // 9. node density
    norms_kernel<<<N_, blk, 0, stream>>>(Xl, norms);
    rho_kernel<<<N_, blk, 0, stream>>>(big, norms, idx, rho_x);
    rho_e_kernel<<<NE / 256, blk, 0, stream>>>(idx, rho_x, rho_e);
    // 10. WXl = Xl@W, WE = E@W
    gemm_wmma_f32<<<grid_nar, blk, 0, stream>>>(Xl, W, WXl, N_, D_, D_, D_, D_, D_, 0, 0);
    gemm_wmma_f32<<<grid_nar, blk, 0, stream>>>(E, W, WE, N_, D_, D_, D_, D_, D_, 0, 0);
    // 11. rank-1 attention vectors + maxes + rho_t + coef denominators
    uv_kernel<<<N_, blk, 0, stream>>>(WXl, WE, ax, ae, u_x, v_x, u_e, v_e);
    max6_kernel<<<6, blk, 0, stream>>>(u_x, v_x, u_e, v_e, rho_x, rho_e, scal);
    rho_t_kernel<<<N_ / 256, blk, 0, stream>>>(rho_x, rho_e, scal, rt_x, rt_e);
    den_kernel<<<N_ / 256, blk, 0, stream>>>(colptr, colidx, u_x, v_x, rt_x,
                                             u_e, v_e, rt_e, den_x, den_e);
    // 12. E_tilde = elu(COE_X.T @ WXl) — COE generated inside the GEMM (no P buffer)
    gemm_attn_wmma<<<grid_nar, blk, 0, stream>>>(u_x, v_x, rt_x, den_x, WXl, Etil,
                                                 N_, D_, N_, D_, D_);
    // 13. X_tilde = elu(COE_E.T @ E_tilde) -> d_out
    gemm_attn_wmma<<<grid_nar, blk, 0, stream>>>(u_e, v_e, rt_e, den_e, Etil, out,
                                                 N_, D_, N_, D_, D_);
}